// GraphConv_38070590111773
// MI455X (gfx1250) — compile-verified
//
#include <hip/hip_runtime.h>
#include <stdint.h>

// GraphConv: out[s,m,f] = sum_c ( sum_n adj[m,n] * x[s,n,c] ) * W[c,f]
// s = b*T+t (768 slices), N=883, C=F=64, adj ~4% sparse, fp32.
//
// Memory-bound problem (~350 MB HBM => ~15us floor @ 23.3 TB/s).
// Strategy: stage x-slice (226 KB) in LDS via async-to-LDS DMA, sparsity-aware
// SpMM on VALU (ballot/readlane gather, ~3.1 GFLOP), dense projection on
// v_wmma_f32_16x16x4_f32 with B-fragments held in registers.

typedef float v2f __attribute__((ext_vector_type(2)));
typedef float v8f __attribute__((ext_vector_type(8)));

#define B_ 64
#define T_ 12
#define N_ 883
#define C_ 64
#define F_ 64
#define SLICES (B_ * T_)
#define NC (N_ * C_)              // 56512 floats per slice (226 KB)
#define TILES ((N_ + 15) / 16)    // 56 row-tiles
#define NWAVES 8
#define AXS_STRIDE 66             // padded row stride: breaks 16-way bank conflict

__global__ void __launch_bounds__(256, 1)
graphconv_kernel(const float* __restrict__ x,
                 const float* __restrict__ adj,
                 const float* __restrict__ w,
                 float* __restrict__ out)
{
    __shared__ __align__(16) float xs[NC];                        // 226048 B: whole x slice
    __shared__ __align__(16) float axs[NWAVES * 16 * AXS_STRIDE]; // 33792 B: per-wave ax tiles

    const int s    = blockIdx.x;
    const int tid  = threadIdx.x;
    const int lane = tid & 31;
    const int wave = tid >> 5;

    const float* xsrc = x   + (size_t)s * NC;
    float*       outp = out + (size_t)s * (N_ * F_);

    // ---- Phase 1: stage x[s] into LDS via async DMA (no VGPR round-trip) ----
    {
        const uint32_t xsbase = (uint32_t)(uintptr_t)&xs[0]; // low 32 bits = LDS offset
        const float4* src4 = reinterpret_cast<const float4*>(xsrc);
        for (int i = tid; i < NC / 4; i += 256) {
            const uint32_t ldsaddr = xsbase + (uint32_t)i * 16u;
            const float4* gaddr = src4 + i;
            asm volatile("global_load_async_to_lds_b128 %0, %1, off"
                         :: "v"(ldsaddr), "v"(gaddr) : "memory");
        }
        asm volatile("s_wait_asynccnt 0" ::: "memory");
    }

    const int mrow = lane & 15;   // M (and N-col) index within 16x16 fragment
    const int hi   = lane >> 4;   // half-wave selector

    // ---- Preload all W B-fragments into registers (tile-invariant) ----
    // bf[jt][k4]: B fragment (4x16 f32) for col-tile jt, K-chunk k4.
    v2f bf[4][16];
    #pragma unroll
    for (int jt = 0; jt < 4; ++jt) {
        const int col = jt * 16 + mrow;
        #pragma unroll
        for (int k4 = 0; k4 < 16; ++k4) {
            const int kb = 4 * k4 + 2 * hi;
            bf[jt][k4].x = w[(kb + 0) * F_ + col];
            bf[jt][k4].y = w[(kb + 1) * F_ + col];
        }
    }

    __syncthreads();   // xs visible to all waves

    float* axw = axs + wave * (16 * AXS_STRIDE);

    for (int t = wave; t < TILES; t += NWAVES) {
        const int m0 = t * 16;
        const bool full = (m0 + 16 <= N_);   // uniform: 55 of 56 tiles are full

        // ---- Step A: sparsity-aware SpMM for 16 rows of ax, into LDS ----
        for (int mi = 0; mi < 16; ++mi) {
            const int m = m0 + mi;
            v2f acc; acc.x = 0.0f; acc.y = 0.0f;   // lane owns c = 2*lane, 2*lane+1
            if (m < N_) {
                const float* arow = adj + (size_t)m * N_;
                if (m + 1 < N_) __builtin_prefetch(arow + N_, 0, 3);
                for (int n0 = 0; n0 < N_; n0 += 32) {
                    const int n  = n0 + lane;
                    const float av = (n < N_) ? arow[n] : 0.0f;
                    unsigned mask = __builtin_amdgcn_ballot_w32(av != 0.0f);
                    while (mask) {
                        const int j = __builtin_ctz(mask);
                        mask &= mask - 1;
                        const float a =
                            __int_as_float(__builtin_amdgcn_readlane(__float_as_int(av), j));
                        const v2f xv =
                            *reinterpret_cast<const v2f*>(&xs[(n0 + j) * C_ + 2 * lane]);
                        acc.x += a * xv.x;
                        acc.y += a * xv.y;
                    }
                }
            }
            *reinterpret_cast<v2f*>(&axw[mi * AXS_STRIDE + 2 * lane]) = acc;
        }

        // Same-wave LDS RAW: make ax stores visible before A-fragment loads.
        asm volatile("s_wait_dscnt 0" ::: "memory");

        // ---- Step B: out_tile[16,64] = ax_tile[16,64] @ W[64,64] via WMMA f32 ----
        #pragma unroll
        for (int jt = 0; jt < 4; ++jt) {
            const int col = jt * 16 + mrow;
            v8f acc = {};
            #pragma unroll
            for (int k4 = 0; k4 < 16; ++k4) {
                const int kb = 4 * k4 + 2 * hi;   // A fragment K pair for this half-wave
                const v2f a = *reinterpret_cast<const v2f*>(&axw[mrow * AXS_STRIDE + kb]);
                acc = __builtin_amdgcn_wmma_f32_16x16x4_f32(
                    /*neg_a=*/false, a, /*neg_b=*/false, bf[jt][k4],
                    /*c_mod=*/(short)0, acc, /*reuse_a=*/false, /*reuse_b=*/false);
            }
            // D layout: VGPR r holds M = r + 8*hi, N = col
            if (full) {
                #pragma unroll
                for (int r = 0; r < 8; ++r)
                    outp[(size_t)(m0 + hi * 8 + r) * F_ + col] = acc[r];
            } else {
                #pragma unroll
                for (int r = 0; r < 8; ++r) {
                    const int m = m0 + hi * 8 + r;
                    if (m < N_) outp[(size_t)m * F_ + col] = acc[r];
                }
            }
        }
    }
}

extern "C" void kernel_launch(void* const* d_in, const int* in_sizes, int n_in,
                              void* d_out, int out_size, void* d_ws, size_t ws_size,
                              hipStream_t stream) {
    const float* x   = (const float*)d_in[0];  // [B,T,N,C]
    const float* adj = (const float*)d_in[1];  // [N,N]
    const float* w   = (const float*)d_in[2];  // [C,F]
    float*       out = (float*)d_out;          // [B,T,N,F]
    (void)in_sizes; (void)n_in; (void)out_size; (void)d_ws; (void)ws_size;

    graphconv_kernel<<<SLICES, 256, 0, stream>>>(x, adj, w, out);
}